// FDTDSimulator_47906065220111
// MI455X (gfx1250) — compile-verified
//
#include <hip/hip_runtime.h>
#include <stdint.h>

// ---------------- problem constants (match reference) ----------------
#define NXG 400
#define NYG 400
#define PMLN 10
#define PORTS 4
#define PORT_W 20
#define SRC_LOC 30
#define DET_LOC 370
#define CSIZE 240
#define COFF 80
#define NCIRC 8
#define STEPS 300
#define COURANT 0.5f
#define PML_SIG 0.5f

// ---------------- decomposition ----------------
#define RPB 8                 // rows per slab (workgroup)
#define NSLAB (NXG / RPB)     // 50 slabs per batch
#define TPB 256               // 8 wave32 waves
#define BATCHN 2
#define NWG (NSLAB * BATCHN)  // 100 workgroups, all co-resident

#define PI_F 3.14159265358979323846f

// ---------------- feature probes ----------------
#if defined(__has_builtin)
#if __has_builtin(__builtin_amdgcn_tensor_load_to_lds)
#define USE_TDM_LD 1
#pragma message("CDNA5 probe: __builtin_amdgcn_tensor_load_to_lds AVAILABLE")
#endif
#if __has_builtin(__builtin_amdgcn_tensor_store_from_lds)
#define USE_TDM_ST 1
#pragma message("CDNA5 probe: __builtin_amdgcn_tensor_store_from_lds AVAILABLE")
#endif
#if __has_builtin(__builtin_amdgcn_wmma_f32_16x16x4_f32)
#define USE_WMMA_DIST 1
#pragma message("CDNA5 probe: __builtin_amdgcn_wmma_f32_16x16x4_f32 AVAILABLE")
#endif
#endif
#ifndef USE_TDM_LD
#define USE_TDM_LD 0
#pragma message("CDNA5 probe: tensor_load_to_lds MISSING - atomic-load fallback")
#endif
#ifndef USE_TDM_ST
#define USE_TDM_ST 0
#pragma message("CDNA5 probe: tensor_store_from_lds MISSING - atomic-store fallback")
#endif
#ifndef USE_WMMA_DIST
#define USE_WMMA_DIST 0
#pragma message("CDNA5 probe: wmma_f32_16x16x4_f32 MISSING - no WMMA aux path")
#endif

typedef unsigned int tdm_u32x4 __attribute__((ext_vector_type(4)));
typedef int tdm_i32x4 __attribute__((ext_vector_type(4)));
typedef int tdm_i32x8 __attribute__((ext_vector_type(8)));
typedef float v2f __attribute__((ext_vector_type(2)));
typedef float v8f __attribute__((ext_vector_type(8)));

// TDM cpol: TH[2:0]=RT(0), SCOPE[4:3]=DEV(2) -> transfers complete device-wide
#define TDM_CPOL (2 << 3)

#if USE_TDM_LD || USE_TDM_ST
// Build D# (CDNA5 ISA) for a contiguous 400x1 tile of
// 4-byte elements.  g0: count=1, lds_addr, global_addr, type=2.
__device__ __forceinline__ tdm_u32x4 tdm_g0(unsigned lds_byte_off,
                                            unsigned long long ga) {
  tdm_u32x4 g0;
  g0[0] = 1u;                                      // count=1 (user descriptor)
  g0[1] = lds_byte_off;                            // lds_addr  [63:32]
  g0[2] = (unsigned)(ga & 0xffffffffull);          // global_addr [95:64]
  g0[3] = ((unsigned)((ga >> 32) & 0x01ffffffull)) // global_addr [120:96]
          | 0x80000000u;                           // type=2 ("image")
  return g0;
}
__device__ __forceinline__ tdm_i32x8 tdm_g1_row400() {
  tdm_i32x8 g1;
  g1[0] = (int)(2u << 16);    // workgroup_mask=0, data_size=2 (4 bytes)
  g1[1] = (int)(400u << 16);  // tensor_dim0.lo16 = 400
  g1[2] = (int)(1u << 16);    // tensor_dim0.hi16=0, tensor_dim1.lo16=1
  g1[3] = (int)(400u << 16);  // tensor_dim1.hi16=0, tile_dim0=400
  g1[4] = 1;                  // tile_dim1=1, tile_dim2=0
  g1[5] = 400;                // tensor_dim0_stride.lo32
  g1[6] = (int)(400u << 16);  // stride0.hi16=0, tensor_dim1_stride.lo16=400
  g1[7] = 0;
  return g1;
}
#endif

#if USE_TDM_LD
__device__ __forceinline__ void tdm_load_row400(unsigned lds_byte_off,
                                                const float* gsrc) {
  tdm_u32x4 g0 = tdm_g0(lds_byte_off, (unsigned long long)(uintptr_t)gsrc);
  tdm_i32x8 g1 = tdm_g1_row400();
  tdm_i32x4 z4 = {0, 0, 0, 0};
#if defined(__clang_major__) && (__clang_major__ >= 23)
  tdm_i32x8 z8 = {0, 0, 0, 0, 0, 0, 0, 0};
  __builtin_amdgcn_tensor_load_to_lds(g0, g1, z4, z4, z8, TDM_CPOL);
#else
  __builtin_amdgcn_tensor_load_to_lds(g0, g1, z4, z4, TDM_CPOL);
#endif
}
#endif

#if USE_TDM_ST
__device__ __forceinline__ void tdm_store_row400(unsigned lds_byte_off,
                                                 float* gdst) {
  tdm_u32x4 g0 = tdm_g0(lds_byte_off, (unsigned long long)(uintptr_t)gdst);
  tdm_i32x8 g1 = tdm_g1_row400();
  tdm_i32x4 z4 = {0, 0, 0, 0};
#if defined(__clang_major__) && (__clang_major__ >= 23)
  tdm_i32x8 z8 = {0, 0, 0, 0, 0, 0, 0, 0};
  __builtin_amdgcn_tensor_store_from_lds(g0, g1, z4, z4, z8, TDM_CPOL);
#else
  __builtin_amdgcn_tensor_store_from_lds(g0, g1, z4, z4, TDM_CPOL);
#endif
}
#endif

// ---------------- helpers ----------------
__device__ __forceinline__ float damp1d(int i) {
  int m = (i < (NXG - 1 - i)) ? i : (NXG - 1 - i);
  if (m < PMLN) {
    float t = ((float)PMLN - (float)m - 0.5f) / (float)PMLN;
    return expf(-PML_SIG * t * t * t);
  }
  return 1.0f;
}

// Exact (reference-identical) permittivity: dist2 is an exact f32 integer,
// compared against fl(r*r) just like jnp does.
__device__ float eps_at(int gi, int gj, const float* sRad) {
  float eps = 1.0f;
  if (gi < SRC_LOC || gi >= DET_LOC) {
#pragma unroll
    for (int p = 0; p < PORTS; ++p) {
      int pc = (p + 1) * (NYG / (PORTS + 1));
      if (gj >= pc - PORT_W / 2 && gj < pc + PORT_W - PORT_W / 2) eps = 2.8f;
    }
  }
  if (gi >= COFF && gi < COFF + CSIZE && gj >= COFF && gj < COFF + CSIZE) {
    float x = (float)(gi - COFF), y = (float)(gj - COFF);
    bool inside = false;
#pragma unroll 4
    for (int k = 0; k < NCIRC * NCIRC; ++k) {
      float r = sRad[k];
      r = (r < 0.3f) ? 0.0f : r;  // reference zeroes tiny radii (keeps <= cmp)
      float cx = 15.0f + 30.0f * (float)(k / NCIRC);
      float cy = 15.0f + 30.0f * (float)(k % NCIRC);
      float dx = x - cx, dy = y - cy;
      inside = inside || (dx * dx + dy * dy <= r * r);
    }
    eps = inside ? 1.0f : 2.8f;
  }
  return eps;
}

// device-wide sense-reversal barrier (all NWG workgroups co-resident)
__device__ __forceinline__ void grid_sync(unsigned* bar) {
  __threadfence();   // release: make halo data agent-visible
  __syncthreads();
  if (threadIdx.x == 0) {
    unsigned* cnt = bar;
    unsigned* gen = bar + 1;
    unsigned g = __hip_atomic_load(gen, __ATOMIC_RELAXED, __HIP_MEMORY_SCOPE_AGENT);
    unsigned a = __hip_atomic_fetch_add(cnt, 1u, __ATOMIC_ACQ_REL, __HIP_MEMORY_SCOPE_AGENT);
    if (a == (unsigned)(NWG - 1)) {
      __hip_atomic_store(cnt, 0u, __ATOMIC_RELAXED, __HIP_MEMORY_SCOPE_AGENT);
      __hip_atomic_fetch_add(gen, 1u, __ATOMIC_ACQ_REL, __HIP_MEMORY_SCOPE_AGENT);
    } else {
      while (__hip_atomic_load(gen, __ATOMIC_ACQUIRE, __HIP_MEMORY_SCOPE_AGENT) == g) {
        __builtin_amdgcn_s_sleep(1);
      }
    }
  }
  __syncthreads();
  __threadfence();   // acquire: invalidate stale near caches before halo reads
}

__global__ void fdtd_init_barrier(unsigned* bar) {
  bar[0] = 0u;
  bar[1] = 0u;
}

// ---------------- persistent FDTD kernel ----------------
// grid = (NSLAB, BATCHN); slab s owns global rows [8s, 8s+8), all 400 columns.
// Fields live entirely in LDS; per step only 4 halo rows per slab cross global
// memory, all moved by the Tensor Data Mover.
__global__ __launch_bounds__(TPB) void fdtd_persistent(
    const float* __restrict__ phases,   // (2, PORTS) in units of pi
    const float* __restrict__ radius,   // (8, 8)
    float* __restrict__ out,            // (2, PORTS, PORT_W)
    float* __restrict__ haloEz,         // [2][NSLAB][NYG]
    float* __restrict__ haloHy,         // [2][NSLAB][NYG]
    float* __restrict__ scratchClear,   // [NWG][8 waves][32] aux clearance
    unsigned* __restrict__ bar) {
  const int slab = blockIdx.x;
  const int batch = blockIdx.y;
  const int tid = threadIdx.x;
  const int r0 = slab * RPB;

  __shared__ float sEz[RPB][NYG];
  __shared__ float sHx[RPB][NYG];
  __shared__ float sHy[RPB][NYG];
  __shared__ float sEInv[RPB][NYG];  // COURANT / eps
  __shared__ float sDampY[NYG];
  __shared__ float sDampX[RPB];
  __shared__ float sEzTop[NYG];      // Ez row r0+RPB from slab+1 (TDM target)
  __shared__ float sHyBot[NYG];      // Hy row r0-1  from slab-1 (TDM target)
  __shared__ float sRad[NCIRC * NCIRC];

  if (tid < NCIRC * NCIRC) sRad[tid] = radius[tid];
  if (tid < RPB) sDampX[tid] = damp1d(r0 + tid);
  for (int c = tid; c < NYG; c += TPB) sDampY[c] = damp1d(c);
  __syncthreads();

  for (int r = 0; r < RPB; ++r)
    for (int c = tid; c < NYG; c += TPB) {
      sEz[r][c] = 0.0f;
      sHx[r][c] = 0.0f;
      sHy[r][c] = 0.0f;
      sEInv[r][c] = COURANT / eps_at(r0 + r, c, sRad);
    }

#if USE_WMMA_DIST
  // ---- auxiliary WMMA pass (setup-only): dist^2 of 16-cell x 16-circle
  // tiles via v_wmma_f32_16x16x4_f32.  dist2 = [x^2+y^2, -2x, -2y, 1] x
  // [1, cx, cy, cx^2+cy^2]^T ; every product/partial sum is an integer
  // < 2^24, so the WMMA result is exact.  The authoritative eps stays on the
  // scalar path above (the 32-bit B operand layout is not hardware-verifiable
  // in a compile-only loop); here we materialize the per-slab minimum signed
  // clearance min(dist2 - r2) into scratch.
  const bool centerSlab = (r0 >= COFF) && (r0 + RPB <= COFF + CSIZE);
  if (centerSlab) {
    const int lane = tid & 31;
    const int wave = tid >> 5;
    const int Ml = lane & 15;
    const bool lo = lane < 16;
    float cmin = 3.0e38f;
    for (int tl = wave; tl < (CSIZE / 16) * RPB; tl += TPB / 32) {
      int r = tl / (CSIZE / 16);
      int cseg = tl - r * (CSIZE / 16);
      float xf = (float)(r0 + r - COFF);
      float yf = (float)(cseg * 16 + Ml);
      // A (16x4, M=lane%16): lanes 0-15 -> K0,K1 ; lanes 16-31 -> K2,K3
      v2f A;
      A[0] = lo ? (xf * xf + yf * yf) : (-2.0f * yf);
      A[1] = lo ? (-2.0f * xf) : 1.0f;
      v8f Cz = {0.f, 0.f, 0.f, 0.f, 0.f, 0.f, 0.f, 0.f};
#pragma unroll
      for (int g = 0; g < 4; ++g) {
        int n = (g << 4) | Ml;  // circle index, N = lane%16
        float cxn = 15.0f + 30.0f * (float)(n >> 3);
        float cyn = 15.0f + 30.0f * (float)(n & 7);
        float rr = sRad[n];
        rr = (rr < 0.3f) ? 0.0f : rr;
        float r2 = rr * rr;
        // B (4x16): lanes 0-15 -> K0,K1 rows; lanes 16-31 -> K2,K3 rows
        v2f B;
        B[0] = lo ? 1.0f : cyn;
        B[1] = lo ? cxn : (cxn * cxn + cyn * cyn);
        v8f D = __builtin_amdgcn_wmma_f32_16x16x4_f32(
            false, A, false, B, (short)0, Cz, false, false);
#pragma unroll
        for (int j = 0; j < 8; ++j) cmin = fminf(cmin, D[j] - r2);
      }
    }
    scratchClear[((batch * NSLAB + slab) * (TPB / 32) + wave) * 32 + lane] =
        cmin;
  }
#else
  (void)scratchClear;
#endif

  float* myHaloEz = haloEz + (batch * NSLAB + slab) * NYG;
  float* myHaloHy = haloHy + (batch * NSLAB + slab) * NYG;
  const float* nextHaloEz = myHaloEz + NYG;  // written by slab+1
  const float* prevHaloHy = myHaloHy - NYG;  // written by slab-1
  for (int c = tid; c < NYG; c += TPB) {
    __hip_atomic_store(myHaloEz + c, 0.0f, __ATOMIC_RELAXED, __HIP_MEMORY_SCOPE_AGENT);
    __hip_atomic_store(myHaloHy + c, 0.0f, __ATOMIC_RELAXED, __HIP_MEMORY_SCOPE_AGENT);
    sEzTop[c] = 0.0f;
    sHyBot[c] = 0.0f;
  }

  const bool hasSrc = (SRC_LOC >= r0) && (SRC_LOC < r0 + RPB);
  const bool hasDet = (DET_LOC >= r0) && (DET_LOC < r0 + RPB);
  float srcPhase = 0.0f;
  int portCol = 0;
  if (tid < PORTS * PORT_W) {
    int p = tid / PORT_W, w = tid - p * PORT_W;
    portCol = (p + 1) * (NYG / (PORTS + 1)) - PORT_W / 2 + w;
    srcPhase = PI_F * phases[batch * PORTS + p];
  }

  grid_sync(bar);  // halos initialized everywhere

  const float OMEGA = 2.0f * PI_F * (12.5f / 1550.0f);  // 2*pi*DT/PERIOD

  for (int t = 0; t < STEPS; ++t) {
    // ---- fetch Ez top halo for this step (TDM: global -> LDS) ----
    if (slab < NSLAB - 1) {
#if USE_TDM_LD
      if (tid < 32) {  // wave 0 issues one TDM op for the whole workgroup
        tdm_load_row400((unsigned)(uintptr_t)&sEzTop[0], nextHaloEz);
        __builtin_amdgcn_s_wait_tensorcnt(0);
      }
#else
      for (int c = tid; c < NYG; c += TPB)
        sEzTop[c] = __hip_atomic_load(nextHaloEz + c, __ATOMIC_RELAXED,
                                      __HIP_MEMORY_SCOPE_AGENT);
#endif
    }
    __syncthreads();

    // ---- H phase: Hx = D*(Hx - 0.5*dEz/dy), Hy = D*(Hy + 0.5*dEz/dx) ----
    for (int r = 0; r < RPB; ++r) {
      float dxr = sDampX[r];
      for (int c = tid; c < NYG; c += TPB) {
        float ez = sEz[r][c];
        float ezR = (c < NYG - 1) ? sEz[r][c + 1] : ez;        // pad 0 at y=end
        float ezD = (r < RPB - 1) ? sEz[r + 1][c]
                                  : ((slab < NSLAB - 1) ? sEzTop[c] : ez);
        float d = dxr * sDampY[c];
        sHx[r][c] = d * (sHx[r][c] - COURANT * (ezR - ez));
        sHy[r][c] = d * (sHy[r][c] + COURANT * (ezD - ez));
      }
    }
    __syncthreads();
    // ---- publish bottom Hy row for slab+1 (TDM: LDS -> global) ----
#if USE_TDM_ST
    if (tid < 32) {
      tdm_store_row400((unsigned)(uintptr_t)&sHy[RPB - 1][0], myHaloHy);
      __builtin_amdgcn_s_wait_tensorcnt(0);
    }
#else
    for (int c = tid; c < NYG; c += TPB)
      __hip_atomic_store(myHaloHy + c, sHy[RPB - 1][c], __ATOMIC_RELAXED,
                         __HIP_MEMORY_SCOPE_AGENT);
#endif
    grid_sync(bar);

    // ---- fetch Hy bottom halo written this step (TDM: global -> LDS) ----
    if (slab > 0) {
#if USE_TDM_LD
      if (tid < 32) {
        tdm_load_row400((unsigned)(uintptr_t)&sHyBot[0], prevHaloHy);
        __builtin_amdgcn_s_wait_tensorcnt(0);
      }
#else
      for (int c = tid; c < NYG; c += TPB)
        sHyBot[c] = __hip_atomic_load(prevHaloHy + c, __ATOMIC_RELAXED,
                                      __HIP_MEMORY_SCOPE_AGENT);
#endif
    }
    __syncthreads();

    // ---- E phase: Ez = D*(Ez + (0.5/eps)*(dHy/dx - dHx/dy)) ----
    for (int r = 0; r < RPB; ++r) {
      float dxr = sDampX[r];
      for (int c = tid; c < NYG; c += TPB) {
        float hy = sHy[r][c];
        float hyU;
        if (r > 0)
          hyU = sHy[r - 1][c];
        else
          hyU = (slab > 0) ? sHyBot[c] : hy;                   // pad 0 at x=0
        float hx = sHx[r][c];
        float hxL = (c > 0) ? sHx[r][c - 1] : hx;              // pad 0 at y=0
        sEz[r][c] = dxr * sDampY[c] *
                    (sEz[r][c] + sEInv[r][c] * ((hy - hyU) - (hx - hxL)));
      }
    }
    __syncthreads();

    // ---- soft line source (after damped E update, as in reference) ----
    if (hasSrc) {
      if (tid < PORTS * PORT_W)
        sEz[SRC_LOC - r0][portCol] += sinf(OMEGA * (float)t + srcPhase);
      __syncthreads();
    }

    // ---- publish top Ez row for slab-1's next H phase (TDM) ----
#if USE_TDM_ST
    if (tid < 32) {
      tdm_store_row400((unsigned)(uintptr_t)&sEz[0][0], myHaloEz);
      __builtin_amdgcn_s_wait_tensorcnt(0);
    }
#else
    for (int c = tid; c < NYG; c += TPB)
      __hip_atomic_store(myHaloEz + c, sEz[0][c], __ATOMIC_RELAXED,
                         __HIP_MEMORY_SCOPE_AGENT);
#endif
    grid_sync(bar);
  }

  // ---- detector: final Ez at the output ports ----
  if (hasDet && tid < PORTS * PORT_W)
    out[batch * PORTS * PORT_W + tid] = sEz[DET_LOC - r0][portCol];
}

// ---------------- host entry ----------------
extern "C" void kernel_launch(void* const* d_in, const int* in_sizes, int n_in,
                              void* d_out, int out_size, void* d_ws,
                              size_t ws_size, hipStream_t stream) {
  (void)in_sizes; (void)n_in; (void)out_size; (void)ws_size;
  const float* phases = (const float*)d_in[0];   // (2, PORTS) float32
  const float* radius = (const float*)d_in[1];   // (8, 8) float32
  float* out = (float*)d_out;                    // (2, PORTS, PORT_W) float32

  // workspace: [0,256) barrier; Ez halos; Hy halos; WMMA clearance scratch
  unsigned* bar = (unsigned*)d_ws;
  float* haloEz = (float*)((char*)d_ws + 256);
  float* haloHy = haloEz + BATCHN * NSLAB * NYG;
  float* scratchClear = haloHy + BATCHN * NSLAB * NYG;

  fdtd_init_barrier<<<1, 1, 0, stream>>>(bar);   // deterministic across replays
  dim3 grid(NSLAB, BATCHN, 1);
  fdtd_persistent<<<grid, TPB, 0, stream>>>(phases, radius, out, haloEz,
                                            haloHy, scratchClear, bar);
}